// Attentive_29703993819978
// MI455X (gfx1250) — compile-verified
//
#include <hip/hip_runtime.h>

// out[i,j] = x[i,j] * w[j]   (ROWS=16384, COLS=4096, f32)
// Bandwidth-bound streaming kernel:
//   - w tile staged once per block via CDNA5 async global->LDS copy (ASYNCcnt)
//   - x streamed with non-temporal b128 loads, out with non-temporal b128 stores
//   - one base address per thread; all 16 rows addressed via constant
//     24-bit immediate offsets (row stride 16 KiB, well within +-8 MiB)

typedef float v4f __attribute__((ext_vector_type(4)));

#define COLS      4096
#define ROWS      16384
#define TPB       256        // 8 wave32s per block
#define COLS_PB   (TPB * 4)  // 1024 columns per block
#define ROWS_PB   16         // rows per block

__global__ __launch_bounds__(TPB) void colscale_stream(
    const float* __restrict__ x,
    const float* __restrict__ w,
    float* __restrict__ out) {
  __shared__ float sw[COLS_PB];

  const unsigned tid     = threadIdx.x;
  const unsigned colBase = blockIdx.x * COLS_PB + tid * 4u;

  // ---- Kick off async stage of this block's w tile into LDS (16B/thread).
  // LDS dest offset = low 32 bits of the flat shared address
  // (ISA 10.2).
  {
    const float* wsrc   = w + colBase;
    unsigned     ldsOff = (unsigned)(unsigned long long)(&sw[tid * 4u]);
    asm volatile("global_load_async_to_lds_b128 %0, %1, off"
                 :
                 : "v"(ldsOff), "v"(wsrc)
                 : "memory");
  }

  // ---- Compute the single x/out base address while the async copy flies.
  const unsigned idx = blockIdx.y * (ROWS_PB * COLS) + colBase;  // < 2^26
  const v4f* __restrict__ px = reinterpret_cast<const v4f*>(x + idx);
  v4f*       __restrict__ po = reinterpret_cast<v4f*>(out + idx);

  // ---- Wait for w, read it back into registers (each thread reads exactly
  // the bytes its own async op wrote; own-wave ASYNCcnt wait suffices).
  asm volatile("s_wait_asynccnt 0" ::: "memory");
  const v4f wv = *reinterpret_cast<const v4f*>(&sw[tid * 4u]);

  // ---- Stream 16 rows: NT b128 load, 4 muls, NT b128 store.
  // Constant pointer offsets -> immediate `offset:` fields, one shared base.
#pragma unroll
  for (int r = 0; r < ROWS_PB; ++r) {
    v4f xv = __builtin_nontemporal_load(px + r * (COLS / 4));
    v4f yv = xv * wv;
    __builtin_nontemporal_store(yv, po + r * (COLS / 4));
  }
}

extern "C" void kernel_launch(void* const* d_in, const int* in_sizes, int n_in,
                              void* d_out, int out_size, void* d_ws, size_t ws_size,
                              hipStream_t stream) {
  const float* x = (const float*)d_in[0];  // (ROWS, COLS) f32
  const float* w = (const float*)d_in[1];  // (COLS,)      f32
  float* out     = (float*)d_out;          // (ROWS, COLS) f32

  dim3 grid(COLS / COLS_PB, ROWS / ROWS_PB);  // (4, 1024) = 4096 blocks
  colscale_stream<<<grid, TPB, 0, stream>>>(x, w, out);
}